// AttnAggregator_8315056685450
// MI455X (gfx1250) — compile-verified
//
#include <hip/hip_runtime.h>

// Problem constants (from reference)
#define B_NODES 20000
#define NNEIGH  32
#define DMODEL  256
#define NHEADS  4
#define DEPTH   64
#define LN_EPS  1e-3f

typedef __attribute__((ext_vector_type(2))) float v2f;
typedef __attribute__((ext_vector_type(4))) float v4f;
typedef __attribute__((ext_vector_type(8))) float v8f;

// LDS row stride: 256 + 4 pad floats -> bank-conflict-free WMMA frag loads
// (bank = (4*m + c) mod 64 distinct across lanes), keeps 16B alignment.
#define XS_STRIDE 260

__global__ __launch_bounds__(256, 1)
void attn_agg_kernel(const float* __restrict__ self_vecs,
                     const float* __restrict__ neigh_vecs,
                     const float* __restrict__ Wq, const float* __restrict__ bq,
                     const float* __restrict__ Wk, const float* __restrict__ bk,
                     const float* __restrict__ Wv, const float* __restrict__ bv,
                     const float* __restrict__ Wd, const float* __restrict__ bd,
                     const float* __restrict__ gamma, const float* __restrict__ beta,
                     float* __restrict__ out_y,   // [B, D]
                     float* __restrict__ out_w)   // [B, H, N]
{
    __shared__ float Xs[NNEIGH * XS_STRIDE];   // neighbor rows      (33.3 KB)
    __shared__ float Ks[NNEIGH * XS_STRIDE];   // K = Xs @ Wk        (33.3 KB)
    __shared__ float Vs[NNEIGH * XS_STRIDE];   // V = Xs @ Wv        (33.3 KB)
    __shared__ float xs[DMODEL];               // self row
    __shared__ float qs[DMODEL];               // q = xs @ Wq
    __shared__ float ws[NHEADS * NNEIGH];      // softmax weights
    __shared__ float ress[DMODEL];             // q + attn
    __shared__ float red[16];                  // LN block reduction

    const int b    = blockIdx.x;
    const int tid  = threadIdx.x;
    const int wave = tid >> 5;
    const int lane = tid & 31;
    const int half = lane >> 4;   // 0: lanes 0-15, 1: lanes 16-31
    const int l16  = lane & 15;

    // ---------------- Stage 1: global -> LDS ----------------
    {
        const float* src = neigh_vecs + (size_t)b * (NNEIGH * DMODEL);
        const int r  = tid >> 3;          // 256 threads -> 32 rows x 8 chunks
        const int c0 = (tid & 7) * 32;
        #pragma unroll
        for (int c = 0; c < 32; c += 4) {
            v4f v = *(const v4f*)(src + r * DMODEL + c0 + c);
            *(v4f*)(&Xs[r * XS_STRIDE + c0 + c]) = v;
        }
        xs[tid] = self_vecs[(size_t)b * DMODEL + tid];
    }
    __syncthreads();

    // ---------------- Stage 2: q = xs @ Wq + bq (thin M=1 -> VALU) ----------------
    {
        float acc = bq[tid];
        #pragma unroll 8
        for (int i = 0; i < DMODEL; ++i)
            acc = fmaf(xs[i], Wq[i * DMODEL + tid], acc);
        qs[tid] = acc;
    }

    // ---------------- Stage 3: K/V projections via V_WMMA_F32_16X16X4_F32 ----------
    // 16 blocks of 32x32 output (8 for K, 8 for V); 8 waves * 2 blocks each.
    {
        for (int blk = 0; blk < 2; ++blk) {
            const int bid = wave * 2 + blk;            // 0..15
            const float* __restrict__ W  = (bid < 8) ? Wk : Wv;
            const float* __restrict__ bi = (bid < 8) ? bk : bv;
            float* __restrict__ Out      = (bid < 8) ? Ks : Vs;
            const int ncol0 = (bid & 7) * 32;

            v8f acc[2][2];
            const v8f z = {};
            acc[0][0] = z; acc[0][1] = z; acc[1][0] = z; acc[1][1] = z;

            for (int k0 = 0; k0 < DMODEL; k0 += 4) {
                const int kr = k0 + half * 2;
                // A frags: lane holds X[m][kr], X[m][kr+1], m = l16 + 16*mtile
                v2f a[2];
                a[0] = *(const v2f*)(&Xs[l16        * XS_STRIDE + kr]);
                a[1] = *(const v2f*)(&Xs[(16 + l16) * XS_STRIDE + kr]);
                // B frags: lane holds W[kr][n], W[kr+1][n], n = ncol0 + 16*ntile + l16
                v2f bf[2];
                #pragma unroll
                for (int j = 0; j < 2; ++j) {
                    const int n = ncol0 + j * 16 + l16;
                    bf[j].x = W[(size_t)kr       * DMODEL + n];
                    bf[j].y = W[(size_t)(kr + 1) * DMODEL + n];
                }
                #pragma unroll
                for (int i = 0; i < 2; ++i)
                    #pragma unroll
                    for (int j = 0; j < 2; ++j)
                        acc[i][j] = __builtin_amdgcn_wmma_f32_16x16x4_f32(
                            false, a[i], false, bf[j], (short)0, acc[i][j],
                            false, false);
            }

            // C/D layout: lanes 0-15 -> M=v, lanes 16-31 -> M=v+8, N=l16
            #pragma unroll
            for (int i = 0; i < 2; ++i)
                #pragma unroll
                for (int j = 0; j < 2; ++j) {
                    const int n = ncol0 + j * 16 + l16;
                    const float bias = bi[n];
                    #pragma unroll
                    for (int v = 0; v < 8; ++v) {
                        const int m = i * 16 + half * 8 + v;
                        Out[m * XS_STRIDE + n] = acc[i][j][v] + bias;
                    }
                }
        }
    }
    __syncthreads();

    // ---------------- Stage 4: scores + pad mask + softmax (waves 0..3 = heads) ----
    if (wave < NHEADS) {
        const int h = wave, n = lane;
        const float* __restrict__ krow = &Ks[n * XS_STRIDE + h * DEPTH];
        const float* __restrict__ qh   = &qs[h * DEPTH];
        float dot = 0.0f;
        int nz = 0;
        #pragma unroll 8
        for (int d = 0; d < DEPTH; ++d) {
            const float kv = krow[d];
            dot = fmaf(qh[d], kv, dot);
            nz |= (kv != 0.0f);
        }
        float score = dot * 0.125f + (nz ? 0.0f : -1e9f);  // 1/sqrt(64)=0.125
        float m = score;
        #pragma unroll
        for (int off = 16; off; off >>= 1) m = fmaxf(m, __shfl_xor(m, off, 32));
        const float e = expf(score - m);
        float s = e;
        #pragma unroll
        for (int off = 16; off; off >>= 1) s += __shfl_xor(s, off, 32);
        const float w = e / s;
        ws[h * NNEIGH + n] = w;
        out_w[(size_t)b * (NHEADS * NNEIGH) + h * NNEIGH + n] = w;
    }
    __syncthreads();

    // ---------------- Stage 5: attn = w @ V ; residual ----------------
    {
        const int h = tid / DEPTH;
        const float* __restrict__ wh = &ws[h * NNEIGH];
        float acc = 0.0f;
        #pragma unroll
        for (int n = 0; n < NNEIGH; ++n)
            acc = fmaf(wh[n], Vs[n * XS_STRIDE + tid], acc);
        ress[tid] = qs[tid] + acc;
    }
    __syncthreads();

    // ---------------- Stage 6: y = res @ Wd + bd ; LayerNorm ; ReLU ----------------
    {
        float y = bd[tid];
        #pragma unroll 8
        for (int i = 0; i < DMODEL; ++i)
            y = fmaf(ress[i], Wd[i * DMODEL + tid], y);

        float s = y, s2 = y * y;
        #pragma unroll
        for (int off = 16; off; off >>= 1) {
            s  += __shfl_xor(s,  off, 32);
            s2 += __shfl_xor(s2, off, 32);
        }
        if (lane == 0) { red[wave] = s; red[8 + wave] = s2; }
        __syncthreads();
        float ts = 0.0f, ts2 = 0.0f;
        #pragma unroll
        for (int i = 0; i < 8; ++i) { ts += red[i]; ts2 += red[8 + i]; }
        const float mu  = ts  * (1.0f / DMODEL);
        const float var = ts2 * (1.0f / DMODEL) - mu * mu;
        const float yn  = (y - mu) * rsqrtf(var + LN_EPS) * gamma[tid] + beta[tid];
        out_y[(size_t)b * DMODEL + tid] = fmaxf(yn, 0.0f);
    }
}

extern "C" void kernel_launch(void* const* d_in, const int* in_sizes, int n_in,
                              void* d_out, int out_size, void* d_ws, size_t ws_size,
                              hipStream_t stream) {
    (void)in_sizes; (void)n_in; (void)out_size; (void)d_ws; (void)ws_size;
    const float* self_vecs = (const float*)d_in[0];
    const float* neigh     = (const float*)d_in[1];
    const float* Wq = (const float*)d_in[2];  const float* bq = (const float*)d_in[3];
    const float* Wk = (const float*)d_in[4];  const float* bk = (const float*)d_in[5];
    const float* Wv = (const float*)d_in[6];  const float* bv = (const float*)d_in[7];
    const float* Wd = (const float*)d_in[8];  const float* bd = (const float*)d_in[9];
    const float* gamma = (const float*)d_in[10];
    const float* beta  = (const float*)d_in[11];

    float* out_y = (float*)d_out;                            // [B, D]
    float* out_w = out_y + (size_t)B_NODES * DMODEL;         // [B, H, 1, N]

    attn_agg_kernel<<<B_NODES, 256, 0, stream>>>(
        self_vecs, neigh, Wq, bq, Wk, bk, Wv, bv, Wd, bd, gamma, beta,
        out_y, out_w);
}